// EMADecomposition_12592844111969
// MI455X (gfx1250) — compile-verified
//
#include <hip/hip_runtime.h>
#include <cstdint>

typedef float        v2f  __attribute__((ext_vector_type(2)));
typedef float        v8f  __attribute__((ext_vector_type(8)));
typedef unsigned int v4u  __attribute__((ext_vector_type(4)));
typedef int          v4i  __attribute__((ext_vector_type(4)));
typedef int          v8i  __attribute__((ext_vector_type(8)));

#define EMA_ALPHA 0.3f
#define EMA_DECAY 0.7f

constexpr int kB = 64, kL = 4096, kC = 128;
constexpr int kNSeg       = 8;
constexpr int kSegLen     = kL / kNSeg;   // 512
constexpr int kChunkT     = 16;
constexpr int kWarmChunks = 4;            // 64 warmup steps; 0.7^64 ~ 1e-10 << fp32 eps

// One TDM descriptor: 2D tile, 16 rows x 128 fp32 (contiguous rows, stride 128)
__device__ __forceinline__ void tdm_load_tile_16x128(const float* gsrc, void* lds_dst) {
  uint64_t ga = (uint64_t)(uintptr_t)gsrc;
  uint32_t lo = (uint32_t)(uintptr_t)lds_dst;       // flat LDS addr low 32 bits = LDS offset
  v4u g0;
  g0.x = 1u;                                        // count=1 (valid user descriptor)
  g0.y = lo;                                        // lds_addr
  g0.z = (uint32_t)ga;                              // global_addr[31:0]
  g0.w = (uint32_t)((ga >> 32) & 0x01FFFFFFu) | (2u << 30);   // global_addr[56:32] | type=2
  v8i g1;
  g1[0] = (int)0x00020000u;                         // data_size = 4 bytes; no pad/iterate/mask
  g1[1] = (int)((uint32_t)kC << 16);                // tensor_dim0 = 128 (lo16 in [31:16])
  g1[2] = 0;                                        // tensor_dim0 hi16 = 0, tensor_dim1 lo16 = 0
  g1[3] = (int)(((uint32_t)kC << 16) | 0x10u);      // tile_dim0 = 128 | tensor_dim1 hi16 (tdim1 = 1<<20)
  g1[4] = kChunkT;                                  // tile_dim1 = 16, tile_dim2 = 0
  g1[5] = kC;                                       // tensor_dim0_stride = 128
  g1[6] = 0;
  g1[7] = 0;
  v4i gz4 = {0, 0, 0, 0};
  v8i gz8 = {0, 0, 0, 0, 0, 0, 0, 0};
  __builtin_amdgcn_tensor_load_to_lds(g0, g1, gz4, gz4, gz8, 0);
}

__global__ __launch_bounds__(256) void ema_decomp_kernel(const float* __restrict__ x,
                                                         float* __restrict__ trend,
                                                         float* __restrict__ seasonal) {
  __shared__ float smem[2][kChunkT][kC];            // 2 x 8KB double buffer

  const int blk  = blockIdx.x;
  const int b    = blk / kNSeg;
  const int seg  = blk % kNSeg;
  const int lane = threadIdx.x & 31;
  const int wv   = threadIdx.x >> 5;                // 8 waves; wave w owns channels [16w,16w+16)
  const int h    = lane >> 4;                       // lane half (WMMA layout split)
  const int col  = lane & 15;
  const int cbase = wv * 16;

  const int warm    = (seg == 0) ? 0 : kWarmChunks;
  const int t0      = seg * kSegLen - warm * kChunkT;
  const int nchunks = kSegLen / kChunkT + warm;

  const float* xb = x + (size_t)b * kL * kC;

  // Per-lane constant A-matrix slices: M[t,k] = a*d^(t-k), k<=t (lower triangular).
  // A 16x4 f32 layout: lanes 0-15 row=lane K={0,1}; lanes 16-31 row=lane-16 K={2,3}.
  v2f Aw[4];
  float dec[8];
  {
    const int row = col;
    for (int s = 0; s < 4; ++s) {
      const int k0 = 4 * s + 2 * h;
      Aw[s].x = (k0     <= row) ? EMA_ALPHA * __powf(EMA_DECAY, (float)(row - k0))     : 0.0f;
      Aw[s].y = (k0 + 1 <= row) ? EMA_ALPHA * __powf(EMA_DECAY, (float)(row - k0 - 1)) : 0.0f;
    }
    for (int r = 0; r < 8; ++r)                     // decay vector d^(t+1), t = r + 8h
      dec[r] = __powf(EMA_DECAY, (float)(r + 8 * h + 1));
  }

  // Initial carry: seg 0 uses c = x[b,0,:] (makes trend[0]=x[0] exact); warmup segs start at 0.
  float carry = 0.0f;
  if (seg == 0) carry = xb[cbase + col];

  if (wv == 0) tdm_load_tile_16x128(xb + (size_t)t0 * kC, &smem[0][0][0]);

  for (int i = 0; i < nchunks; ++i) {
    const int tch = t0 + i * kChunkT;

    if (wv == 0) {
      if (i + 1 < nchunks) {
        tdm_load_tile_16x128(xb + (size_t)(tch + kChunkT) * kC, &smem[(i + 1) & 1][0][0]);
        __builtin_amdgcn_s_wait_tensorcnt(1);       // chunk i landed (TDM in-order per wave)
      } else {
        __builtin_amdgcn_s_wait_tensorcnt(0);
      }
    }
    __syncthreads();

    const float (*tile)[kC] = smem[i & 1];

    // B operand slices (4x16 f32): lane half h, VGPR j -> element (t = 4s+2h+j, c)
    v2f Bw[4];
    for (int s = 0; s < 4; ++s) {
      const int t = 4 * s + 2 * h;
      Bw[s].x = tile[t][cbase + col];
      Bw[s].y = tile[t + 1][cbase + col];
    }

    v8f acc = {};
    acc = __builtin_amdgcn_wmma_f32_16x16x4_f32(false, Aw[0], false, Bw[0], (short)0, acc, false, false);
    acc = __builtin_amdgcn_wmma_f32_16x16x4_f32(false, Aw[1], false, Bw[1], (short)0, acc, false, false);
    acc = __builtin_amdgcn_wmma_f32_16x16x4_f32(false, Aw[2], false, Bw[2], (short)0, acc, false, false);
    acc = __builtin_amdgcn_wmma_f32_16x16x4_f32(false, Aw[3], false, Bw[3], (short)0, acc, false, false);

    // trend = M@x + d^(t+1) * carry
    float tr[8];
    for (int r = 0; r < 8; ++r) tr[r] = acc[r] + dec[r] * carry;

    // next carry = trend row 15 = VGPR7 of lanes 16..31; broadcast per channel
    carry = __shfl(tr[7], 16 + col, 32);

    if (i >= warm) {                                // uniform branch; EXEC stays full
      const size_t obase = (size_t)b * kL * kC + (size_t)tch * kC + cbase + col;
      for (int r = 0; r < 8; ++r) {
        const int   tl = r + 8 * h;                 // C/D layout row for this VGPR
        const float xv = tile[tl][cbase + col];
        const size_t o = obase + (size_t)tl * kC;
        trend[o]    = tr[r];
        seasonal[o] = xv - tr[r];
      }
    }
    __syncthreads();                                // protect buffer (i&1) before reuse at i+2
  }
}

extern "C" void kernel_launch(void* const* d_in, const int* in_sizes, int n_in,
                              void* d_out, int out_size, void* d_ws, size_t ws_size,
                              hipStream_t stream) {
  (void)in_sizes; (void)n_in; (void)d_ws; (void)ws_size; (void)out_size;
  const float* x = (const float*)d_in[0];
  float* trend    = (float*)d_out;
  float* seasonal = trend + (size_t)kB * kL * kC;
  dim3 grid(kB * kNSeg);
  dim3 block(256);
  ema_decomp_kernel<<<grid, block, 0, stream>>>(x, trend, seasonal);
}